// HyperConnectionTorch_26525718020213
// MI455X (gfx1250) — compile-verified
//
#include <hip/hip_runtime.h>
#include <hip/hip_bf16.h>
#include <math.h>

// Problem constants (fixed by setup_inputs)
#define BATCH   4
#define SEQ     4096
#define NSTR    4
#define DIM     1024
#define IN_DIM  (NSTR * DIM)              // 4096
#define OUT24   (NSTR * NSTR + 2 * NSTR)  // 24
#define EPS     1e-6f
#define SINKH   20

typedef float __attribute__((ext_vector_type(2))) v2f;
typedef float __attribute__((ext_vector_type(4))) f4;
typedef float __attribute__((ext_vector_type(8))) v8f;

// ---------------------------------------------------------------------------
// Kernel 1: partial column sums of H over s-chunks (float4 lanes).
// H layout: [b][s][n*d], inner dim IN_DIM contiguous.
// grid = (IN_DIM/4/256 = 4, 16 chunks, BATCH); block = 256
// partials (as f4): [(chunk*BATCH + b)*(IN_DIM/4) + jv]
// ---------------------------------------------------------------------------
__global__ void hc_colsum_kernel(const float* __restrict__ H,
                                 float* __restrict__ partials) {
    const int jv    = blockIdx.x * 256 + threadIdx.x;   // f4 column 0..1023
    const int chunk = blockIdx.y;
    const int b     = blockIdx.z;
    const f4* p = reinterpret_cast<const f4*>(H) +
                  ((size_t)b * SEQ + (size_t)chunk * 256) * (IN_DIM / 4) + jv;
    f4 acc = {0.0f, 0.0f, 0.0f, 0.0f};
#pragma unroll 8
    for (int s = 0; s < 256; ++s) {
        acc += __builtin_nontemporal_load(p + (size_t)s * (IN_DIM / 4));
    }
    reinterpret_cast<f4*>(partials)[((size_t)chunk * BATCH + b) * (IN_DIM / 4) + jv] = acc;
}

// ---------------------------------------------------------------------------
// Kernel 2: combine 16 partials -> x = mean over s.  grid = 16, block = 256 (f4)
// ---------------------------------------------------------------------------
__global__ void hc_colsum2_kernel(const float* __restrict__ partials,
                                  float* __restrict__ x) {
    const int idx = blockIdx.x * 256 + threadIdx.x;   // 0..4095 (f4 units)
    const f4* p4 = reinterpret_cast<const f4*>(partials);
    f4 acc = {0.0f, 0.0f, 0.0f, 0.0f};
#pragma unroll
    for (int c = 0; c < 16; ++c) {
        acc += p4[(size_t)c * (BATCH * IN_DIM / 4) + idx];
    }
    reinterpret_cast<f4*>(x)[idx] = acc * (1.0f / (float)SEQ);
}

// ---------------------------------------------------------------------------
// Kernel 3: fused dynamic-weight computation.  One block, 256 threads (8 waves).
//   Phase A: inv_rms per batch (LDS tree reduction of sum(x^2)).
//   Phase B: raw = x @ phi via V_WMMA_F32_16X16X4_F32.
//            Padding rows/cols are handled by ADDRESS CLAMPING, not zeroing:
//            garbage A rows 4..15 only affect D rows 4..15 and garbage B cols
//            24..31 only affect D cols 24..31 -- never read.  Inner loop is
//            branch-free: pure loads + 2 WMMAs, EXEC all-ones throughout.
//   Phase C: sigmoid / normalize / sinkhorn (threads 0..3), emit W[b] (5x5).
// ---------------------------------------------------------------------------
__global__ void hc_weights_kernel(const float* __restrict__ x,
                                  const float* __restrict__ phi,
                                  const float* __restrict__ pre_base,
                                  const float* __restrict__ post_base,
                                  const float* __restrict__ res_base,
                                  const float* __restrict__ alpha_pre,
                                  const float* __restrict__ alpha_post,
                                  const float* __restrict__ alpha_res,
                                  float* __restrict__ wmat) {
    __shared__ float red[256];
    __shared__ float invr[BATCH];
    __shared__ float rawPart[8][BATCH * OUT24];
    __shared__ float rawS[BATCH * OUT24];

    const int t = threadIdx.x;

    // ---- Phase A: sum of squares of x per batch -> inv_rms ----
    {
        const int b = t >> 6;          // 4 groups of 64 threads
        const int g = t & 63;
        const f4* x4 = reinterpret_cast<const f4*>(x) + b * (IN_DIM / 4);
        float acc = 0.0f;
#pragma unroll
        for (int j = g; j < IN_DIM / 4; j += 64) {
            f4 v = x4[j];
            acc = fmaf(v.x, v.x, acc);
            acc = fmaf(v.y, v.y, acc);
            acc = fmaf(v.z, v.z, acc);
            acc = fmaf(v.w, v.w, acc);
        }
        red[t] = acc;
        __syncthreads();
        for (int off = 32; off > 0; off >>= 1) {
            if (g < off) red[t] += red[t + off];
            __syncthreads();
        }
        if (g == 0) invr[b] = 1.0f / sqrtf(red[t] * (1.0f / (float)IN_DIM) + EPS);
        __syncthreads();
    }

    // ---- Phase B: raw = x @ phi using WMMA f32 16x16x4 (all 8 waves) ----
    {
        const int wave = t >> 5;
        const int lane = t & 31;
        const int m    = lane & 15;          // A row (batch) / B column
        const int ko   = (lane >> 4) << 1;   // lanes 0-15 -> K+{0,1}; 16-31 -> K+{2,3}
        const int mrow = m & 3;              // clamped A row (garbage rows unused)
        const int bhiC = 16 + (m & 7);       // clamped high B col (garbage cols unused)

        v8f c0 = {};   // output columns 0..15
        v8f c1 = {};   // output columns 16..23 (24..31 garbage, unused)

        const float* xr = x + mrow * IN_DIM;
        const int kk0 = wave * 512;
#pragma unroll 4
        for (int kk = kk0; kk < kk0 + 512; kk += 4) {
            const int k0 = kk + ko;
            const int k1 = k0 + 1;
            v2f a  = { xr[k0], xr[k1] };                       // adjacent -> b64
            v2f b0 = { phi[k0 * OUT24 + m],    phi[k1 * OUT24 + m] };
            v2f b1 = { phi[k0 * OUT24 + bhiC], phi[k1 * OUT24 + bhiC] };
            c0 = __builtin_amdgcn_wmma_f32_16x16x4_f32(false, a, false, b0,
                                                       (short)0, c0, false, false);
            c1 = __builtin_amdgcn_wmma_f32_16x16x4_f32(false, a, false, b1,
                                                       (short)0, c1, false, false);
        }

        // C layout: lanes 0-15 hold N=lane; VGPR v holds M=v (v=0..7).
        if (lane < 16) {
            rawPart[wave][0 * OUT24 + lane] = c0[0];
            rawPart[wave][1 * OUT24 + lane] = c0[1];
            rawPart[wave][2 * OUT24 + lane] = c0[2];
            rawPart[wave][3 * OUT24 + lane] = c0[3];
            if (lane < OUT24 - 16) {
                rawPart[wave][0 * OUT24 + 16 + lane] = c1[0];
                rawPart[wave][1 * OUT24 + 16 + lane] = c1[1];
                rawPart[wave][2 * OUT24 + 16 + lane] = c1[2];
                rawPart[wave][3 * OUT24 + 16 + lane] = c1[3];
            }
        }
        __syncthreads();
        if (t < BATCH * OUT24) {
            float s = 0.0f;
#pragma unroll
            for (int w = 0; w < 8; ++w) s += rawPart[w][t];
            rawS[t] = s;
        }
        __syncthreads();
    }

    // ---- Phase C: sigmoid / normalize / sinkhorn, write W[b] (5x5) ----
    if (t < BATCH) {
        const int b = t;
        const float iR    = invr[b];
        const float aPre  = alpha_pre[0];
        const float aPost = alpha_post[0];
        const float aRes  = alpha_res[0];
        const float* raw  = &rawS[b * OUT24];

        float pre[NSTR], post[NSTR], P[NSTR][NSTR];
        float sumpre = 0.0f;
#pragma unroll
        for (int n = 0; n < NSTR; ++n) {
            float sp = iR * aPre * raw[n] + pre_base[n];
            pre[n] = 1.0f / (1.0f + expf(-sp));
            sumpre += pre[n];
        }
        const float inv_sum = 1.0f / (sumpre + EPS);
#pragma unroll
        for (int n = 0; n < NSTR; ++n) pre[n] *= inv_sum;
#pragma unroll
        for (int n = 0; n < NSTR; ++n) {
            float sp = iR * aPost * raw[NSTR + n] + post_base[n];
            post[n] = 2.0f / (1.0f + expf(-sp));
        }
#pragma unroll
        for (int i = 0; i < NSTR; ++i)
#pragma unroll
            for (int j = 0; j < NSTR; ++j) {
                float sr = iR * aRes * raw[2 * NSTR + i * NSTR + j] +
                           res_base[i * NSTR + j];
                P[i][j] = fabsf(sr) + EPS;
            }
#pragma unroll 1
        for (int it = 0; it < SINKH; ++it) {
#pragma unroll
            for (int i = 0; i < NSTR; ++i) {
                float rs = P[i][0] + P[i][1] + P[i][2] + P[i][3] + EPS;
                float ir = 1.0f / rs;
#pragma unroll
                for (int j = 0; j < NSTR; ++j) P[i][j] *= ir;
            }
#pragma unroll
            for (int j = 0; j < NSTR; ++j) {
                float cs = P[0][j] + P[1][j] + P[2][j] + P[3][j] + EPS;
                float ic = 1.0f / cs;
#pragma unroll
                for (int i = 0; i < NSTR; ++i) P[i][j] *= ic;
            }
        }
        float* W = wmat + b * 25;    // 5 rows x 5 cols
#pragma unroll
        for (int k = 0; k < NSTR; ++k) W[k] = pre[k];
        W[4] = 0.0f;
#pragma unroll
        for (int i = 0; i < NSTR; ++i) {
#pragma unroll
            for (int j = 0; j < NSTR; ++j) W[(1 + i) * 5 + j] = P[i][j];
            W[(1 + i) * 5 + 4] = post[i];
        }
    }
}

// ---------------------------------------------------------------------------
// Kernel 4: streaming 5x5 stream mix.  One (b,s) per block; thread = float4 of d.
// out[b][s][j][:] = sum_k W[j][k] * {H streams, branch}[k][:]
// grid = BATCH*SEQ = 16384, block = 256
// ---------------------------------------------------------------------------
__global__ void hc_mix_kernel(const float* __restrict__ H,
                              const float* __restrict__ branch,
                              const float* __restrict__ wmat,
                              float* __restrict__ out) {
    const int bs = blockIdx.x;           // b*SEQ + s
    const int c  = threadIdx.x;          // float4 index within d (0..255)
    const int b  = bs >> 12;             // / SEQ

    // Block-uniform weight loads -> scalar path
    const float* W = wmat + b * 25;
    float w[25];
#pragma unroll
    for (int i = 0; i < 25; ++i) w[i] = W[i];

    const f4* Hp = reinterpret_cast<const f4*>(H) + (size_t)bs * (NSTR * DIM / 4) + c;
    f4 h0 = __builtin_nontemporal_load(Hp);
    f4 h1 = __builtin_nontemporal_load(Hp + (DIM / 4));
    f4 h2 = __builtin_nontemporal_load(Hp + 2 * (DIM / 4));
    f4 h3 = __builtin_nontemporal_load(Hp + 3 * (DIM / 4));
    f4 bo = __builtin_nontemporal_load(
        reinterpret_cast<const f4*>(branch) + (size_t)bs * (DIM / 4) + c);

    f4* O = reinterpret_cast<f4*>(out) + (size_t)bs * (5 * DIM / 4) + c;
#pragma unroll
    for (int j = 0; j < 5; ++j) {
        f4 r = w[j * 5 + 0] * h0 + w[j * 5 + 1] * h1 + w[j * 5 + 2] * h2 +
               w[j * 5 + 3] * h3 + w[j * 5 + 4] * bo;
        __builtin_nontemporal_store(r, O + j * (DIM / 4));
    }
}

// ---------------------------------------------------------------------------
extern "C" void kernel_launch(void* const* d_in, const int* in_sizes, int n_in,
                              void* d_out, int out_size, void* d_ws, size_t ws_size,
                              hipStream_t stream) {
    const float* H      = (const float*)d_in[0];
    const float* branch = (const float*)d_in[1];
    const float* phi    = (const float*)d_in[2];
    const float* pre_b  = (const float*)d_in[3];
    const float* post_b = (const float*)d_in[4];
    const float* res_b  = (const float*)d_in[5];
    const float* a_pre  = (const float*)d_in[6];
    const float* a_post = (const float*)d_in[7];
    const float* a_res  = (const float*)d_in[8];
    float* out = (float*)d_out;

    float* ws       = (float*)d_ws;
    float* partials = ws;                                  // 16*4*4096 = 262144 f
    float* x        = ws + 16 * BATCH * IN_DIM;            // 16384 f
    float* wmat     = x + BATCH * IN_DIM;                  // 100 f

    hc_colsum_kernel<<<dim3(IN_DIM / 4 / 256, 16, BATCH), 256, 0, stream>>>(H, partials);
    hc_colsum2_kernel<<<(BATCH * IN_DIM / 4) / 256, 256, 0, stream>>>(partials, x);
    hc_weights_kernel<<<1, 256, 0, stream>>>(x, phi, pre_b, post_b, res_b,
                                             a_pre, a_post, a_res, wmat);
    hc_mix_kernel<<<BATCH * SEQ, 256, 0, stream>>>(H, branch, wmat, out);
}